// CrossLevelQuadConv_16458314678336
// MI455X (gfx1250) — compile-verified
//
#include <hip/hip_runtime.h>
#include <stdint.h>

typedef float v2f __attribute__((ext_vector_type(2)));
typedef float v8f __attribute__((ext_vector_type(8)));

#define MAXD        8
#define CHN         128          // channels per feature row
#define OUTC        128          // output channels
#define NSLOT       9
#define NROWS       65536        // 4^8
#define KREF        1161         // 9*(C+1), reference K
#define KMAIN       1152         // 9*C, WMMA K
#define ROWS_PER_WG 64
#define THREADS     256
#define ASTRIDE     132          // padded LDS row stride (floats)
#define ASZ         (ROWS_PER_WG * ASTRIDE)

// ---------------- prep: repack W into WMMA-B-friendly layout in workspace ----
// wt2 layout: for k in [0,1152), o in [0,128):
//   wt2[(k>>1)*256 + o*2 + (k&1)] = W[o*1161 + (k/128)*129 + (k%128)]
// so a lane's B fragment {B[k][o], B[k+1][o]} (k even) is one contiguous b64.
__global__ __launch_bounds__(256) void prep_wt2(const float* __restrict__ W,
                                                float* __restrict__ wt2) {
  int g   = blockIdx.x * 256 + threadIdx.x;   // [0, 576*256) == 1152*128
  int p   = g >> 8;
  int rem = g & 255;
  int o   = rem >> 1;
  int par = rem & 1;
  int k   = (p << 1) + par;
  int s   = k >> 7;
  int c   = k & 127;
  wt2[g] = W[o * KREF + s * 129 + c];
}

// delta-channel weights wd[s*128+o] = W[o*1161 + s*129 + 128], then a 128-float
// zero page (gather target for invalid neighbors) immediately after.
__global__ __launch_bounds__(256) void prep_wd(const float* __restrict__ W,
                                               float* __restrict__ wd) {
  int i = blockIdx.x * 256 + threadIdx.x;     // 5*256 = 1280 == 1152 + 128
  if (i < NSLOT * OUTC) {
    int s = i >> 7;
    int o = i & 127;
    wd[i] = W[o * KREF + s * 129 + 128];
  } else if (i < NSLOT * OUTC + CHN) {
    wd[i] = 0.0f;                             // zero page
  }
}

// issue one 16-byte async global->LDS copy; IOFFSET applies to both addresses
#define ASYNC16(OFF)                                                        \
  asm volatile("global_load_async_to_lds_b128 %0, %1, off offset:" #OFF    \
               :: "v"(ldsA), "v"(ga) : "memory")

// ---------------- main fused gather + WMMA GEMM kernel ----------------------
__global__ __launch_bounds__(THREADS) void quadconv_main(
    const float* __restrict__ f0, const float* __restrict__ f1,
    const float* __restrict__ f2, const float* __restrict__ f3,
    const float* __restrict__ f4, const float* __restrict__ f5,
    const float* __restrict__ f6, const float* __restrict__ f7,
    const float* __restrict__ f8,
    const int*   __restrict__ nidx, const int* __restrict__ ndep,
    const float* __restrict__ wt2,  const float* __restrict__ wdelta,
    const float* __restrict__ bvec, const int* __restrict__ depthPtr,
    const float* __restrict__ zpage,
    float* __restrict__ out)
{
  __shared__ float              As[2 * ASZ];                    // double-buffered A tile
  __shared__ unsigned long long addrT[ROWS_PER_WG * NSLOT];     // gather byte addresses
  __shared__ float              deltaT[ROWS_PER_WG * NSLOT];    // depth-delta channel
  __shared__ float              WdLds[NSLOT * OUTC + OUTC];     // delta weights + bias
  __shared__ unsigned long long featP[NSLOT];                   // per-depth base pointers

  const int tid     = threadIdx.x;
  const int lane    = tid & 31;
  const int wave    = tid >> 5;
  const int rowBase = blockIdx.x * ROWS_PER_WG;
  const int depth   = depthPtr[0];

  if (tid < NSLOT) {
    const float* p;
    switch (tid) {
      case 0: p = f0; break; case 1: p = f1; break; case 2: p = f2; break;
      case 3: p = f3; break; case 4: p = f4; break; case 5: p = f5; break;
      case 6: p = f6; break; case 7: p = f7; break; default: p = f8; break;
    }
    featP[tid] = (unsigned long long)p;
  }
  for (int i = tid; i < NSLOT * OUTC + OUTC; i += THREADS)
    WdLds[i] = (i < NSLOT * OUTC) ? wdelta[i] : bvec[i - NSLOT * OUTC];
  __syncthreads();

  // Gather addresses / delta for all 64 rows x 9 slots.
  // Invalid neighbors point at the zero page (matches reference g * valid).
  for (int e = tid; e < ROWS_PER_WG * NSLOT; e += THREADS) {
    int r = e / NSLOT;
    int s = e - r * NSLOT;
    int n = rowBase + r;
    int idx = nidx[n * NSLOT + s];
    int dep = ndep[n * NSLOT + s];
    bool valid = (idx != -1);
    int dcl  = dep < 0 ? 0 : (dep > MAXD ? MAXD : dep);
    int sz   = 1 << (2 * dcl);
    int idxc = idx < 0 ? 0 : (idx > sz - 1 ? sz - 1 : idx);
    addrT[e]  = valid ? featP[dcl] + (unsigned long long)idxc * (CHN * 4)
                      : (unsigned long long)zpage;
    deltaT[e] = valid ? (float)(depth - dep) * 0.125f : 0.0f;
  }
  __syncthreads();

  const int n0       = wave * 16;                 // this wave's output-column tile
  const int rA       = lane & 15;                 // A-frag row (M)
  const int koffA    = (lane >> 4) << 1;          // A-frag K offset (0 or 2)
  const int bLaneOff = ((lane >> 4) << 8) | ((lane & 15) << 1);

  const int rStage = tid >> 2;                    // staging: 4 lanes per row
  const int cStage = (tid & 3) << 5;              // 32 floats (128 B) per lane

  v8f acc[4] = {};                                // 4 M-tiles of 16x16 f32

  // prologue: async-stage slot 0 into buffer 0
  {
    const unsigned long long ga = addrT[rStage * NSLOT + 0] + (unsigned)(cStage * 4);
    const uint32_t ldsA = (uint32_t)(uintptr_t)&As[rStage * ASTRIDE + cStage];
    ASYNC16(0);  ASYNC16(16); ASYNC16(32); ASYNC16(48);
    ASYNC16(64); ASYNC16(80); ASYNC16(96); ASYNC16(112);
  }
  asm volatile("s_wait_asynccnt 0x0" ::: "memory");
  __syncthreads();

  for (int s = 0; s < NSLOT; ++s) {
    const float* cur = &As[(s & 1) * ASZ];

    // issue async gather for next slot into the other buffer (overlaps WMMA)
    if (s + 1 < NSLOT) {
      const unsigned long long ga =
          addrT[rStage * NSLOT + (s + 1)] + (unsigned)(cStage * 4);
      const uint32_t ldsA =
          (uint32_t)(uintptr_t)&As[((s + 1) & 1) * ASZ + rStage * ASTRIDE + cStage];
      ASYNC16(0);  ASYNC16(16); ASYNC16(32); ASYNC16(48);
      ASYNC16(64); ASYNC16(80); ASYNC16(96); ASYNC16(112);
    }

    // ---- 128-deep K for this slot: 32 chunks of K=4, 4 M-tiles each ----
    const float* bBase = wt2 + s * (64 * 256) + n0 * 2 + bLaneOff;
    #pragma unroll
    for (int c32 = 0; c32 < 32; ++c32) {
      v2f bfr = *(const v2f*)(bBase + c32 * 512);
      const int k0 = c32 * 4;
      #pragma unroll
      for (int mt = 0; mt < 4; ++mt) {
        v2f afr = *(const v2f*)&cur[(mt * 16 + rA) * ASTRIDE + k0 + koffA];
        acc[mt] = __builtin_amdgcn_wmma_f32_16x16x4_f32(
            /*neg_a=*/false, afr, /*neg_b=*/false, bfr,
            /*c_mod=*/(short)0, acc[mt], /*reuse_a=*/false, /*reuse_b=*/false);
      }
    }

    // next slot's async stores must be complete (and this buffer's reads done)
    asm volatile("s_wait_asynccnt 0x0" ::: "memory");
    __syncthreads();
  }

  // ---- epilogue: bias + rank-9 delta-channel update, store ----
  const int col  = n0 + (lane & 15);
  const int rhal = (lane >> 4) << 3;
  #pragma unroll
  for (int mt = 0; mt < 4; ++mt) {
    #pragma unroll
    for (int j = 0; j < 8; ++j) {
      int rloc = mt * 16 + j + rhal;
      float v = acc[mt][j] + WdLds[NSLOT * OUTC + col];
      #pragma unroll
      for (int s = 0; s < NSLOT; ++s)
        v += deltaT[rloc * NSLOT + s] * WdLds[s * OUTC + col];
      out[(size_t)(rowBase + rloc) * OUTC + col] = v;
    }
  }
}

extern "C" void kernel_launch(void* const* d_in, const int* in_sizes, int n_in,
                              void* d_out, int out_size, void* d_ws, size_t ws_size,
                              hipStream_t stream) {
  const float* f[9];
  for (int d = 0; d < 9; ++d) f[d] = (const float*)d_in[d];
  const int*   nidx     = (const int*)d_in[9];
  const int*   ndep     = (const int*)d_in[10];
  const float* W        = (const float*)d_in[11];
  const float* b        = (const float*)d_in[12];
  const int*   depthPtr = (const int*)d_in[13];

  float* wt2 = (float*)d_ws;                  // 1152*128 floats = 576 KB
  float* wd  = wt2 + KMAIN * OUTC;            // 9*128 delta weights
  float* zp  = wd + NSLOT * OUTC;             // 128-float zero page

  prep_wt2<<<576, 256, 0, stream>>>(W, wt2);
  prep_wd<<<5, 256, 0, stream>>>(W, wd);
  quadconv_main<<<NROWS / ROWS_PER_WG, THREADS, 0, stream>>>(
      f[0], f[1], f[2], f[3], f[4], f[5], f[6], f[7], f[8],
      nidx, ndep, wt2, wd, b, depthPtr, zp, (float*)d_out);
}